// SAGE_24747601559698
// MI455X (gfx1250) — compile-verified
//
#include <hip/hip_runtime.h>
#include <math.h>

#define NN0 585728
#define NN1 22528
#define NN2 2048
#define EE0 563200
#define EE1 20480
#define F_IN 128
#define F_HID 256
#define F_OUT 64

typedef __attribute__((ext_vector_type(2))) float v2f;
typedef __attribute__((ext_vector_type(8))) float v8f;

// ---------------------------------------------------------------------------
// Zero a region of workspace (float4-granular).
// ---------------------------------------------------------------------------
__global__ void zero_kernel(float4* __restrict__ p, int n4) {
    int i = blockIdx.x * blockDim.x + threadIdx.x;
    if (i < n4) p[i] = make_float4(0.f, 0.f, 0.f, 0.f);
}

// ---------------------------------------------------------------------------
// Edge aggregation: one wave32 per edge. Each lane moves a float4 slice of
// the source feature row (coalesced) and scatter-adds with native
// global_atomic_add_f32 (agg target fits in L2). Lane 0 bumps the degree.
// ---------------------------------------------------------------------------
__global__ void agg_kernel(const float* __restrict__ xsrc,
                           const int*   __restrict__ src,
                           const int*   __restrict__ dst,
                           float* __restrict__ agg,
                           float* __restrict__ cnt,
                           int nedges, int F) {
    int gid  = blockIdx.x * blockDim.x + threadIdx.x;
    int wave = gid >> 5;
    int lane = threadIdx.x & 31;
    if (wave >= nedges) return;
    int s = src[wave];
    int d = dst[wave];
    const float* xs = xsrc + (size_t)s * F;
    float*       ag = agg  + (size_t)d * F;
    for (int f = lane * 4; f < F; f += 128) {
        float4 v = *(const float4*)(xs + f);
        atomicAdd(ag + f + 0, v.x);
        atomicAdd(ag + f + 1, v.y);
        atomicAdd(ag + f + 2, v.z);
        atomicAdd(ag + f + 3, v.w);
    }
    if (lane == 0) atomicAdd(cnt + d, 1.0f);
}

// ---------------------------------------------------------------------------
// Fused SAGE linear layer via V_WMMA_F32_16X16X4_F32 (wave32), N-blocked.
//   Out[M,N] = act( mean(Aagg)[M,K] @ Wl[K,N] + Xtgt[M,K] @ Wr[K,N] + bias )
//
// Block = 256 threads = 8 waves; each block owns 8 M-tiles (128 rows) and a
// 64-wide N stripe (blockIdx.y). Weights are staged in LDS *transposed*
// ([col][k], k-stride padded to 68 words) in 64-deep K slabs, so each lane's
// (k,k+1) B pair is one aligned ds_load_b64 into an even VGPR pair — no
// register shuffles — and the k+2/k+3 half-wave hits disjoint banks
// (lane i uses banks {4i,4i+1}, half-wave 2 uses {4i+2,4i+3}: all 64 banks).
//
// Per k-step a wave loads A once per path (b64, mean-normalization folded in)
// and issues 8 WMMAs, path-split so each acc[tn] has distance-4 reuse.
//
// Lane layouts per CDNA5 ISA 7.12.2:
//   A 16x4 f32 : lanes 0-15 -> M=lane, v0=K0 v1=K1 ; lanes 16-31 -> K2/K3
//   B 4x16 f32 : lanes 0-15 -> N=lane, v0=K0 v1=K1 ; lanes 16-31 -> K2/K3
//   C/D 16x16  : VGPR i -> rows i (lanes 0-15) / i+8 (lanes 16-31), col=lane%16
// M % 128 == 0, K % 64 == 0, N % 64 == 0 at all call sites.
// ---------------------------------------------------------------------------
#define KC    64          // K slab depth staged in LDS
#define NPW   64          // N stripe width per block / per wave
#define KPAD  68          // padded LDS k-stride (64 + 4) -> conflict-free b64

__global__ void sage_gemm_kernel(const float* __restrict__ Aagg,
                                 const float* __restrict__ cnt,
                                 const float* __restrict__ Xtgt,
                                 const float* __restrict__ Wl,
                                 const float* __restrict__ Wr,
                                 const float* __restrict__ bias,
                                 float* __restrict__ Out,
                                 int M, int K, int N, int do_relu) {
    __shared__ float lds_l[NPW][KPAD];   // [col][k], transposed slab
    __shared__ float lds_r[NPW][KPAD];

    const int tid    = threadIdx.x;
    const int wv     = tid >> 5;            // wave in block: 0..7
    const int lane   = tid & 31;
    const int lane15 = lane & 15;
    const int khalf  = (lane >> 4) * 2;     // lanes 16-31 carry K+2, K+3

    const int tileM = blockIdx.x * 8 + wv;  // 16-row tile owned by this wave
    const int nb    = blockIdx.y * NPW;     // N stripe base

    const int rowA = tileM * 16 + lane15;
    const float inv = 1.0f / fmaxf(cnt[rowA], 1.0f);

    const float* Arow = Aagg + (size_t)rowA * K;
    const float* Xrow = Xtgt + (size_t)rowA * K;

    v8f acc[4];
#pragma unroll
    for (int tn = 0; tn < 4; ++tn)
        acc[tn] = (v8f){0.f, 0.f, 0.f, 0.f, 0.f, 0.f, 0.f, 0.f};

    for (int kc = 0; kc < K; kc += KC) {
        __syncthreads();
        // Stage the 64x64 weight slabs transposed: lds[cc][kk] = W[kc+kk][nb+cc].
        // Global reads stay fully coalesced (consecutive tid -> consecutive cc).
        for (int i = tid; i < KC * NPW; i += 256) {
            int kk = i >> 6;
            int cc = i & 63;
            lds_l[cc][kk] = Wl[(size_t)(kc + kk) * N + nb + cc];
            lds_r[cc][kk] = Wr[(size_t)(kc + kk) * N + nb + cc];
        }
        __syncthreads();

#pragma unroll 4
        for (int k = 0; k < KC; k += 4) {
            const int kg = kc + k + khalf;
            v2f a1 = *(const v2f*)(Arow + kg);
            a1 = a1 * inv;                       // mean-normalize neighbor path
            v2f a2 = *(const v2f*)(Xrow + kg);
            const int kl = k + khalf;
            // Path 1: neighbor-mean x Wl (A operand held across 4 WMMAs)
#pragma unroll
            for (int tn = 0; tn < 4; ++tn) {
                v2f b = *(const v2f*)&lds_l[tn * 16 + lane15][kl];
                acc[tn] = __builtin_amdgcn_wmma_f32_16x16x4_f32(
                    false, a1, false, b, (short)0, acc[tn], false, false);
            }
            // Path 2: self x Wr
#pragma unroll
            for (int tn = 0; tn < 4; ++tn) {
                v2f b = *(const v2f*)&lds_r[tn * 16 + lane15][kl];
                acc[tn] = __builtin_amdgcn_wmma_f32_16x16x4_f32(
                    false, a2, false, b, (short)0, acc[tn], false, false);
            }
        }
    }

    const int rbase = tileM * 16 + ((lane >> 4) << 3);   // rows rbase..rbase+7
#pragma unroll
    for (int tn = 0; tn < 4; ++tn) {
        const int col = nb + tn * 16 + lane15;
        const float bb = bias[col];
#pragma unroll
        for (int i = 0; i < 8; ++i) {
            float v = acc[tn][i] + bb;
            if (do_relu) v = fmaxf(v, 0.0f);
            Out[(size_t)(rbase + i) * N + col] = v;
        }
    }
}

// ---------------------------------------------------------------------------
// Row-wise log_softmax over N=64 columns: one wave per row, 2 elems/lane,
// shuffle reductions for max and sum(exp).
// ---------------------------------------------------------------------------
__global__ void log_softmax_kernel(float* __restrict__ out, int M, int N) {
    int gid  = blockIdx.x * blockDim.x + threadIdx.x;
    int wave = gid >> 5;
    int lane = threadIdx.x & 31;
    if (wave >= M) return;
    float* row = out + (size_t)wave * N;
    float v0 = row[lane];
    float v1 = row[lane + 32];
    float m = fmaxf(v0, v1);
    for (int off = 16; off > 0; off >>= 1) m = fmaxf(m, __shfl_xor(m, off, 32));
    float s = expf(v0 - m) + expf(v1 - m);
    for (int off = 16; off > 0; off >>= 1) s += __shfl_xor(s, off, 32);
    float lse = m + logf(s);
    row[lane]      = v0 - lse;
    row[lane + 32] = v1 - lse;
}

// ---------------------------------------------------------------------------
extern "C" void kernel_launch(void* const* d_in, const int* in_sizes, int n_in,
                              void* d_out, int out_size, void* d_ws, size_t ws_size,
                              hipStream_t stream) {
    const float* x    = (const float*)d_in[0];
    const float* Wl1  = (const float*)d_in[1];
    const float* Wr1  = (const float*)d_in[2];
    const float* b1   = (const float*)d_in[3];
    const float* Wl2  = (const float*)d_in[4];
    const float* Wr2  = (const float*)d_in[5];
    const float* b2   = (const float*)d_in[6];
    const int*   src0 = (const int*)d_in[7];
    const int*   dst0 = (const int*)d_in[8];
    const int*   src1 = (const int*)d_in[9];
    const int*   dst1 = (const int*)d_in[10];
    float* out = (float*)d_out;

    // Workspace layout (floats):
    //   agg0 [N1*F_IN] | cnt0 [N1] | agg1 [N2*F_HID] | cnt1 [N2] | h [N1*F_HID]
    float* ws   = (float*)d_ws;
    float* agg0 = ws;
    float* cnt0 = agg0 + (size_t)NN1 * F_IN;
    float* agg1 = cnt0 + NN1;
    float* cnt1 = agg1 + (size_t)NN2 * F_HID;
    float* h    = cnt1 + NN2;

    const int BLK = 256;

    // 1) Zero accumulators (contiguous region; count divisible by 4).
    size_t zf = (size_t)NN1 * F_IN + NN1 + (size_t)NN2 * F_HID + NN2;
    int n4 = (int)(zf / 4);
    zero_kernel<<<(n4 + BLK - 1) / BLK, BLK, 0, stream>>>((float4*)ws, n4);

    // 2) Layer-0 edge aggregation: one wave per edge.
    {
        long long threads = (long long)EE0 * 32;
        int blocks = (int)((threads + BLK - 1) / BLK);
        agg_kernel<<<blocks, BLK, 0, stream>>>(x, src0, dst0, agg0, cnt0, EE0, F_IN);
    }

    // 3) Layer-0 linear + bias + ReLU -> h [N1, F_HID]   (WMMA f32, N-blocked)
    {
        dim3 grid(NN1 / 128, F_HID / NPW);   // (176, 4)
        sage_gemm_kernel<<<grid, BLK, 0, stream>>>(agg0, cnt0, x, Wl1, Wr1, b1,
                                                   h, NN1, F_IN, F_HID, 1);
    }

    // 4) Layer-1 edge aggregation over h.
    {
        long long threads = (long long)EE1 * 32;
        int blocks = (int)((threads + BLK - 1) / BLK);
        agg_kernel<<<blocks, BLK, 0, stream>>>(h, src1, dst1, agg1, cnt1, EE1, F_HID);
    }

    // 5) Layer-1 linear + bias -> out [N2, F_OUT]        (WMMA f32, N-blocked)
    {
        dim3 grid(NN2 / 128, F_OUT / NPW);   // (16, 1)
        sage_gemm_kernel<<<grid, BLK, 0, stream>>>(agg1, cnt1, h, Wl2, Wr2, b2,
                                                   out, NN2, F_HID, F_OUT, 0);
    }

    // 6) log_softmax in place on out.
    {
        long long threads = (long long)NN2 * 32;
        int blocks = (int)((threads + BLK - 1) / BLK);
        log_softmax_kernel<<<blocks, BLK, 0, stream>>>(out, NN2, F_OUT);
    }
}